// ID_2585570312402
// MI455X (gfx1250) — compile-verified
//
#include <hip/hip_runtime.h>
#include <math.h>

typedef float v2f __attribute__((ext_vector_type(2)));
typedef float v8f __attribute__((ext_vector_type(8)));

#define NI    128          // instances
#define CDIM  512          // rows (C)
#define LDIM  128          // embedding length
#define EPSV  1e-12f
#define BR    64           // macro tile edge (rows / cols)
#define NSTRIP (CDIM / BR) // 8 strips
#define LDSS  132          // padded LDS row stride (dwords): bank-conflict-free b64 frags

__device__ __forceinline__ float waveRed(float v) {
#pragma unroll
  for (int off = 16; off; off >>= 1) v += __shfl_xor(v, off, 32);
  return v;
}

// ---------------- prep: masked per-row squared norms --------------------
__global__ __launch_bounds__(256) void prep_kernel(
    const float* __restrict__ S, const float* __restrict__ T,
    const int* __restrict__ targets,
    float* __restrict__ sqs, float* __restrict__ sqt) {
  const int wave = (blockIdx.x * blockDim.x + threadIdx.x) >> 5; // one wave per row
  const int lane = threadIdx.x & 31;
  if (wave >= NI * CDIM) return;
  const size_t base = (size_t)wave * LDIM + lane * 4;
  float4 sv = *reinterpret_cast<const float4*>(S + base);
  float4 tv = *reinterpret_cast<const float4*>(T + base);
  float ss = sv.x * sv.x + sv.y * sv.y + sv.z * sv.z + sv.w * sv.w;
  float ts = tv.x * tv.x + tv.y * tv.y + tv.z * tv.z + tv.w * tv.w;
  ss = waveRed(ss);
  ts = waveRed(ts);
  if (lane == 0) {
    const float m = (float)targets[wave];
    sqs[wave] = m * ss;
    sqt[wave] = m * ts;
  }
}

// ---------------- main WMMA pass (MODE 0: distance sums, MODE 1: Huber) -
// Symmetry: D, pm and the Huber term are symmetric; diagonal is zero.
// Block (n, ib) processes macro-blocks cj = ib..7 only; off-diagonal macro
// blocks are weighted x2 (folded into the pair mask).
template <int MODE>
__global__ __launch_bounds__(256) void rkd_pass(
    const float* __restrict__ S, const float* __restrict__ T,
    const int* __restrict__ targets,
    const float* __restrict__ sqs, const float* __restrict__ sqt,
    const float* __restrict__ invms, const float* __restrict__ invmt,
    float* __restrict__ outA, float* __restrict__ outB) {
  __shared__ __attribute__((aligned(16))) float AsS[BR * LDSS];
  __shared__ __attribute__((aligned(16))) float AsT[BR * LDSS];
  __shared__ __attribute__((aligned(16))) float BsS[BR * LDSS];
  __shared__ __attribute__((aligned(16))) float BsT[BR * LDSS];
  __shared__ float sqsI[BR], sqtI[BR], mIs[BR];
  __shared__ float sqsJ[BR], sqtJ[BR], mJs[BR];
  __shared__ float red[16];

  const int n   = blockIdx.x / NSTRIP;
  const int ib  = blockIdx.x % NSTRIP;
  const int I0  = ib * BR;
  const int tid = threadIdx.x;
  const int lane = tid & 31;
  const int w    = tid >> 5;

  const float* __restrict__ Sn = S + (size_t)n * CDIM * LDIM;
  const float* __restrict__ Tn = T + (size_t)n * CDIM * LDIM;
  const int*   __restrict__ Mn = targets + n * CDIM;

  // ---- stage A strip (rows I0..I0+63), masked ----
#pragma unroll
  for (int i = 0; i < 8; ++i) {
    int f = i * 1024 + tid * 4;
    int r = f >> 7, col = f & 127;
    float m = (float)Mn[I0 + r];
    float4 v = *reinterpret_cast<const float4*>(Sn + (size_t)(I0 + r) * LDIM + col);
    v.x *= m; v.y *= m; v.z *= m; v.w *= m;
    *reinterpret_cast<float4*>(&AsS[r * LDSS + col]) = v;
    float4 u = *reinterpret_cast<const float4*>(Tn + (size_t)(I0 + r) * LDIM + col);
    u.x *= m; u.y *= m; u.z *= m; u.w *= m;
    *reinterpret_cast<float4*>(&AsT[r * LDSS + col]) = u;
  }
  if (tid < BR) {
    sqsI[tid] = sqs[n * CDIM + I0 + tid];
    sqtI[tid] = sqt[n * CDIM + I0 + tid];
    mIs[tid]  = (float)Mn[I0 + tid];
  }

  // wave -> subtiles: rows sr (0..3), columns {sc0, sc0+2}
  const int sr   = w & 3;
  const int sc0  = w >> 2;           // 0 or 1
  const int m16  = lane & 15;
  const int hi   = lane >> 4;
  const int koff = hi << 1;

  float accA = 0.f, accB = 0.f;
  float ims = 0.f, imt = 0.f;
  if (MODE == 1) { ims = invms[n]; imt = invmt[n]; }

  const float* apS  = &AsS[(sr * 16 + m16) * LDSS + koff];
  const float* apT  = &AsT[(sr * 16 + m16) * LDSS + koff];
  const float* bpS0 = &BsS[(sc0 * 16 + m16) * LDSS + koff];
  const float* bpS1 = &BsS[((sc0 + 2) * 16 + m16) * LDSS + koff];
  const float* bpT0 = &BsT[(sc0 * 16 + m16) * LDSS + koff];
  const float* bpT1 = &BsT[((sc0 + 2) * 16 + m16) * LDSS + koff];

  for (int cj = ib; cj < NSTRIP; ++cj) {
    __syncthreads(); // A staged / previous B fully consumed
    const int J0 = cj * BR;
    const float wgt = (cj == ib) ? 1.0f : 2.0f; // symmetric double-count
#pragma unroll
    for (int i = 0; i < 8; ++i) {
      int f = i * 1024 + tid * 4;
      int r = f >> 7, col = f & 127;
      float m = (float)Mn[J0 + r];
      float4 v = *reinterpret_cast<const float4*>(Sn + (size_t)(J0 + r) * LDIM + col);
      v.x *= m; v.y *= m; v.z *= m; v.w *= m;
      *reinterpret_cast<float4*>(&BsS[r * LDSS + col]) = v;
      float4 u = *reinterpret_cast<const float4*>(Tn + (size_t)(J0 + r) * LDIM + col);
      u.x *= m; u.y *= m; u.z *= m; u.w *= m;
      *reinterpret_cast<float4*>(&BsT[r * LDSS + col]) = u;
    }
    if (tid < BR) {
      sqsJ[tid] = sqs[n * CDIM + J0 + tid];
      sqtJ[tid] = sqt[n * CDIM + J0 + tid];
      mJs[tid]  = (float)Mn[J0 + tid];
    }
    __syncthreads();

    v8f cs0 = {}; v8f cs1 = {}; v8f ct0 = {}; v8f ct1 = {};
#pragma unroll 4
    for (int k = 0; k < LDIM; k += 4) {
      v2f aS  = *reinterpret_cast<const v2f*>(apS + k);
      v2f bS0 = *reinterpret_cast<const v2f*>(bpS0 + k);
      v2f bS1 = *reinterpret_cast<const v2f*>(bpS1 + k);
      cs0 = __builtin_amdgcn_wmma_f32_16x16x4_f32(false, aS, false, bS0, (short)0, cs0, false, false);
      cs1 = __builtin_amdgcn_wmma_f32_16x16x4_f32(false, aS, false, bS1, (short)0, cs1, false, false);
      v2f aT  = *reinterpret_cast<const v2f*>(apT + k);
      v2f bT0 = *reinterpret_cast<const v2f*>(bpT0 + k);
      v2f bT1 = *reinterpret_cast<const v2f*>(bpT1 + k);
      ct0 = __builtin_amdgcn_wmma_f32_16x16x4_f32(false, aT, false, bT0, (short)0, ct0, false, false);
      ct1 = __builtin_amdgcn_wmma_f32_16x16x4_f32(false, aT, false, bT1, (short)0, ct1, false, false);
    }

    // epilogue: C/D layout  M = r + 8*hi (per VGPR r), N = lane&15
    const int jl0 = sc0 * 16 + m16;
    const int jl1 = (sc0 + 2) * 16 + m16;
    const float sj0s = sqsJ[jl0], sj0t = sqtJ[jl0], mj0 = wgt * mJs[jl0];
    const float sj1s = sqsJ[jl1], sj1t = sqtJ[jl1], mj1 = wgt * mJs[jl1];
#pragma unroll
    for (int r = 0; r < 8; ++r) {
      const int iloc = sr * 16 + r + 8 * hi;
      const int gi   = I0 + iloc;
      const float sis = sqsI[iloc], sit = sqtI[iloc], mi = mIs[iloc];
      {
        float ds = sqrtf(fmaxf(sis + sj0s - 2.0f * cs0[r], EPSV));
        float dt = sqrtf(fmaxf(sit + sj0t - 2.0f * ct0[r], EPSV));
        if (gi == (J0 + jl0)) { ds = 0.f; dt = 0.f; }
        const float pm = mi * mj0; // includes symmetry weight
        if (MODE == 0) { accA += ds * pm; accB += dt * pm; }
        else {
          float diff = ds * ims - dt * imt;
          float ad = fabsf(diff);
          float hub = (ad < 1.f) ? 0.5f * diff * diff : (ad - 0.5f);
          accA += hub * pm;
        }
      }
      {
        float ds = sqrtf(fmaxf(sis + sj1s - 2.0f * cs1[r], EPSV));
        float dt = sqrtf(fmaxf(sit + sj1t - 2.0f * ct1[r], EPSV));
        if (gi == (J0 + jl1)) { ds = 0.f; dt = 0.f; }
        const float pm = mi * mj1; // includes symmetry weight
        if (MODE == 0) { accA += ds * pm; accB += dt * pm; }
        else {
          float diff = ds * ims - dt * imt;
          float ad = fabsf(diff);
          float hub = (ad < 1.f) ? 0.5f * diff * diff : (ad - 0.5f);
          accA += hub * pm;
        }
      }
    }
  }

  accA = waveRed(accA);
  if (MODE == 0) accB = waveRed(accB);
  if (lane == 0) { red[w] = accA; red[8 + w] = accB; }
  __syncthreads();
  if (tid == 0) {
    float a = 0.f, b = 0.f;
    for (int i = 0; i < 8; ++i) { a += red[i]; b += red[8 + i]; }
    outA[blockIdx.x] = a;
    if (MODE == 0) outB[blockIdx.x] = b;
  }
}

// ---------------- per-instance stats: k, means -> inverses + scale -------
__global__ void stats_kernel(const int* __restrict__ targets,
                             const float* __restrict__ sumDsPart,
                             const float* __restrict__ sumDtPart,
                             float* __restrict__ invms, float* __restrict__ invmt,
                             float* __restrict__ scale) {
  const int n = blockIdx.x;
  const int lane = threadIdx.x;
  float k = 0.f;
  for (int c = lane; c < CDIM; c += 32) k += (float)targets[n * CDIM + c];
  k = waveRed(k);
  if (lane == 0) {
    float sds = 0.f, sdt = 0.f;
    for (int i = 0; i < NSTRIP; ++i) {
      sds += sumDsPart[n * NSTRIP + i];
      sdt += sumDtPart[n * NSTRIP + i];
    }
    const float cnt = fmaxf(k * k - k, 1.0f); // = sum(pos): off-diag dists always > 0
    const bool ok = (k > 1.0f);
    invms[n] = ok ? cnt / sds : 0.0f;
    invmt[n] = ok ? cnt / sdt : 0.0f;
    scale[n] = ok ? 1.0f / (k * k) : 0.0f;
  }
}

// ---------------- final scaled reduction --------------------------------
__global__ __launch_bounds__(256) void final_kernel(
    const float* __restrict__ hubPart, const float* __restrict__ scale,
    float* __restrict__ out) {
  __shared__ float red[8];
  const int tid = threadIdx.x;
  float acc = 0.f;
  for (int idx = tid; idx < NI * NSTRIP; idx += 256)
    acc += hubPart[idx] * scale[idx / NSTRIP];
  acc = waveRed(acc);
  if ((tid & 31) == 0) red[tid >> 5] = acc;
  __syncthreads();
  if (tid == 0) {
    float s = 0.f;
    for (int i = 0; i < 8; ++i) s += red[i];
    out[0] = s;
  }
}

extern "C" void kernel_launch(void* const* d_in, const int* in_sizes, int n_in,
                              void* d_out, int out_size, void* d_ws, size_t ws_size,
                              hipStream_t stream) {
  const float* S = (const float*)d_in[0];       // le_student [N,C,L] f32
  const float* T = (const float*)d_in[1];       // le_teacher [N,C,L] f32
  const int* targets = (const int*)d_in[2];     // [N,C] i32

  float* ws = (float*)d_ws;
  float* sqs       = ws;                         // N*C
  float* sqt       = sqs + NI * CDIM;            // N*C
  float* sumDsPart = sqt + NI * CDIM;            // N*NSTRIP
  float* sumDtPart = sumDsPart + NI * NSTRIP;    // N*NSTRIP
  float* hubPart   = sumDtPart + NI * NSTRIP;    // N*NSTRIP
  float* invms     = hubPart + NI * NSTRIP;      // N
  float* invmt     = invms + NI;                 // N
  float* scale     = invmt + NI;                 // N

  prep_kernel<<<(NI * CDIM) / 8, 256, 0, stream>>>(S, T, targets, sqs, sqt);
  rkd_pass<0><<<NI * NSTRIP, 256, 0, stream>>>(S, T, targets, sqs, sqt,
                                               nullptr, nullptr, sumDsPart, sumDtPart);
  stats_kernel<<<NI, 32, 0, stream>>>(targets, sumDsPart, sumDtPart, invms, invmt, scale);
  rkd_pass<1><<<NI * NSTRIP, 256, 0, stream>>>(S, T, targets, sqs, sqt,
                                               invms, invmt, hubPart, nullptr);
  final_kernel<<<1, 256, 0, stream>>>(hubPart, scale, (float*)d_out);
}